// ModelWithAuxiliaryHead_3496103379640
// MI455X (gfx1250) — compile-verified
//
#include <hip/hip_runtime.h>
#include <stdint.h>

#define NSEG   64
#define HIDDEN 512
#define MLP    1024
#define T      512

#define KC     64                      // K-chunk (floats) staged per async batch
#define LPAD   68                      // LDS row pitch in floats (272B: 16B aligned, 4-bank row shift)
#define BUF_FLOATS  (16 * LPAD)        // one buffer: 16 rows
#define WAVE_FLOATS (2 * BUF_FLOATS)   // double buffered
#define SMEM_FLOATS (4 * WAVE_FLOATS)  // 4 waves per block (34,816 B)

typedef __attribute__((ext_vector_type(2))) float v2f;
typedef __attribute__((ext_vector_type(8))) float v8f;

// ---------------------------------------------------------------------------
// Kernel 1: deterministic (atomic-free) grouping of tokens by segment.
// ---------------------------------------------------------------------------
__global__ void group_tokens_kernel(const int* __restrict__ seg_ids,
                                    int* __restrict__ counts,
                                    int* __restrict__ offsets,
                                    int* __restrict__ tok) {
    __shared__ int s_seg[T];
    int t = threadIdx.x;
    int s = seg_ids[t];
    s_seg[t] = s;
    if (t < NSEG) { counts[t] = 0; offsets[t] = 0; }
    __syncthreads();
    int less = 0, rank = 0, cnt = 0;
    for (int i = 0; i < T; ++i) {
        int si = s_seg[i];
        less += (si < s) ? 1 : 0;
        cnt  += (si == s) ? 1 : 0;
        rank += ((si == s) && (i < t)) ? 1 : 0;
    }
    tok[less + rank] = t;
    if (rank == 0) { counts[s] = cnt; offsets[s] = less; }
}

// ---------------------------------------------------------------------------
// Issue one 16-row x 64-float weight chunk into LDS with 8 async b128 loads.
// Instruction i covers rows 2i+half; lanes 0..15 fetch consecutive float4s
// (fully coalesced 256B bursts). gLane/ldsLane are this lane's bases.
// ---------------------------------------------------------------------------
template <int KDIM>
__device__ __forceinline__ void issue_async_chunk(const float* gLane, unsigned ldsLane) {
    // Close the WAR hazard: our previous ds_load reads of this buffer must be
    // complete before the async engine overwrites it.
    asm volatile("s_wait_dscnt 0x0" ::: "memory");
#pragma unroll
    for (int i = 0; i < 8; ++i) {
        const float* g = gLane + (size_t)(2 * i) * KDIM;
        unsigned l = ldsLane + (unsigned)(2 * i) * (LPAD * 4);
        asm volatile("global_load_async_to_lds_b128 %0, %1, off"
                     :: "v"(l), "v"(g) : "memory");
    }
}

// ---------------------------------------------------------------------------
// Grouped GEMM: Out[t, n] = act( A[t,:] . W[s,n,:] + Bias[s,n] )
// One wave owns a 16-wide output tile; fp32 WMMA 16x16x4 over K.
// Weights streamed through double-buffered LDS via async global->LDS.
// ---------------------------------------------------------------------------
template <int KDIM, bool RELU>
__global__ void grouped_gemm_kernel(const float* __restrict__ Arows,  // [T, KDIM]
                                    const float* __restrict__ W,      // [NSEG, ndim, KDIM]
                                    const float* __restrict__ Bias,   // [NSEG, ndim]
                                    const int* __restrict__ counts,
                                    const int* __restrict__ offsets,
                                    const int* __restrict__ tok,
                                    float* __restrict__ Out,          // [T, ndim]
                                    int ndim) {
    __shared__ float smem[SMEM_FLOATS];

    const int s = blockIdx.x;
    const int n = counts[s];
    if (n == 0) return;
    const int base = offsets[s];
    const int lane = threadIdx.x & 31;
    const int wave = threadIdx.x >> 5;
    const int half = lane >> 4;
    const int lo   = lane & 15;
    const int m0   = (blockIdx.y * 4 + wave) * 16;

    const float* wseg = W + (size_t)s * ndim * KDIM;
    const float  bias = Bias[s * ndim + m0 + lo];

    float* myLds = smem + wave * WAVE_FLOATS;
    const unsigned ldsBase = (unsigned)(uintptr_t)myLds;

    // Async-load lane bases: row m0+half, byte column lo*16
    const float* gLane   = wseg + (size_t)(m0 + half) * KDIM + lo * 4;
    const unsigned ldsLane = ldsBase + half * (LPAD * 4) + lo * 16;

    // B-fragment read row for this lane (row index = lo)
    const float* ldsRow0 = myLds + lo * LPAD;
    const float* ldsRow1 = myLds + BUF_FLOATS + lo * LPAD;

    const int nchunk = KDIM / KC;
    const int ntiles = (n + 15) >> 4;

    for (int tt = 0; tt < ntiles; ++tt) {
        int arow = tt * 16 + lo;
        int tA = tok[base + (arow < n ? arow : 0)];
        const float* xrow = Arows + (size_t)tA * KDIM;

        v8f acc = {};
        issue_async_chunk<KDIM>(gLane, ldsLane);     // prime buffer 0
        int buf = 0;
        for (int c = 0; c < nchunk; ++c) {
            if (c + 1 < nchunk) {
                issue_async_chunk<KDIM>(gLane + (c + 1) * KC,
                                        ldsLane + (unsigned)((buf ^ 1) * (BUF_FLOATS * 4)));
                // 16 outstanding, in-order completion: <=8 left means the
                // current buffer's 8 loads have landed.
                asm volatile("s_wait_asynccnt 0x8" ::: "memory");
            } else {
                asm volatile("s_wait_asynccnt 0x0" ::: "memory");
            }
            const float* brow = buf ? ldsRow1 : ldsRow0;
            const float* ar   = xrow + c * KC;
#pragma unroll
            for (int kk = 0; kk < KC; kk += 4) {
                v2f a = *(const v2f*)(ar + kk + 2 * half);       // global (L2-hot)
                v2f b = *(const v2f*)(brow + kk + 2 * half);     // LDS, conflict-free
                acc = __builtin_amdgcn_wmma_f32_16x16x4_f32(
                    false, a, false, b, (short)0, acc, false, false);
            }
            buf ^= 1;
        }

#pragma unroll
        for (int r = 0; r < 8; ++r) {
            int trow = tt * 16 + r + 8 * half;
            if (trow < n) {
                int tg = tok[base + trow];
                float v = acc[r] + bias;
                if (RELU) v = v > 0.f ? v : 0.f;
                Out[(size_t)tg * ndim + m0 + lo] = v;
            }
        }
    }
}

extern "C" void kernel_launch(void* const* d_in, const int* in_sizes, int n_in,
                              void* d_out, int out_size, void* d_ws, size_t ws_size,
                              hipStream_t stream) {
    const float* X   = (const float*)d_in[0];   // [T, HIDDEN]
    const int*   seg = (const int*)d_in[1];     // [T]
    const float* W1  = (const float*)d_in[2];   // [NSEG, MLP, HIDDEN]
    const float* W2  = (const float*)d_in[3];   // [NSEG, HIDDEN, MLP]
    const float* b1  = (const float*)d_in[4];   // [NSEG, MLP]
    const float* b2  = (const float*)d_in[5];   // [NSEG, HIDDEN]
    float* out = (float*)d_out;                 // [T, HIDDEN]

    int*   counts  = (int*)d_ws;
    int*   offsets = counts + NSEG;
    int*   tokidx  = offsets + NSEG;
    float* inter   = (float*)((char*)d_ws + 4096);   // [T, MLP] = 2 MB

    group_tokens_kernel<<<1, T, 0, stream>>>(seg, counts, offsets, tokidx);

    dim3 blk(128);                         // 4 waves
    dim3 g1(NSEG, MLP / 16 / 4);           // 64 x 16
    grouped_gemm_kernel<HIDDEN, true><<<g1, blk, 0, stream>>>(
        X, W1, b1, counts, offsets, tokidx, inter, MLP);

    dim3 g2(NSEG, HIDDEN / 16 / 4);        // 64 x 8
    grouped_gemm_kernel<MLP, false><<<g2, blk, 0, stream>>>(
        inter, W2, b2, counts, offsets, tokidx, out, HIDDEN);
}